// SparseCrossModalAlign_85942295593047
// MI455X (gfx1250) — compile-verified
//
#include <hip/hip_runtime.h>
#include <math.h>

// ---------------------------------------------------------------------------
// Fused sparse cross-modal attention for MI455X (gfx1250, wave32, WMMA).
// One block per m=(b,n) head: QKV projection + QK^T + top-16 softmax + AV +
// residual LayerNorm, all from LDS, f32 WMMA (V_WMMA_F32_16X16X4_F32).
// Memory-bound problem (~350MB @ 23.3TB/s), so full fusion + f32 is optimal.
// ---------------------------------------------------------------------------

typedef float v2f __attribute__((ext_vector_type(2)));
typedef float v8f __attribute__((ext_vector_type(8)));

#define ZB 4
#define ZS 128
#define ZN 512
#define ZH 64
#define TOPK 16

#define ST  68    // padded LDS row stride (floats) for 64-wide tiles (17 words, odd -> no bank conflicts)
#define STA 132   // padded LDS row stride for 128-wide attn tile (33 words, odd)

// LDS layout (in floats)
#define OFF_X    0                      // 128*68 traffic rows (lives until epilogue)
#define OFF_Q    (OFF_X + ZS*ST)
#define OFF_K    (OFF_Q + ZS*ST)
#define OFF_V    (OFF_K + ZS*ST)
#define OFF_RB   (OFF_V + ZS*ST)        // region B: { sY + sW }  union  { sAttn }
#define OFF_Y    OFF_RB                 // 128*68 text rows     (phases 0-1 only)
#define OFF_W    (OFF_Y + ZS*ST)        // 3 * 64*68 weights    (phases 0-1 only)
#define OFF_ATTN OFF_RB                 // 128*132 attn probs   (phases 3-4)
#define REGB_SZ  (ZS*ST + 3*ZH*ST)      // 21760 >= 128*132 = 16896
#define OFF_CONST (OFF_RB + REGB_SZ)    // 3*64 biases + 64 gamma + 64 beta
#define LDS_FLOATS (OFF_CONST + 3*ZH + ZH + ZH)

__global__ __launch_bounds__(256, 1)
void sparse_xmodal_fused_kernel(const float* __restrict__ traffic,
                                const float* __restrict__ text,
                                const float* __restrict__ Wq, const float* __restrict__ bq,
                                const float* __restrict__ Wk, const float* __restrict__ bk,
                                const float* __restrict__ Wv, const float* __restrict__ bv,
                                const float* __restrict__ gamma, const float* __restrict__ beta,
                                float* __restrict__ out, float* __restrict__ attn_out)
{
    extern __shared__ float smem[];
    const int tid    = threadIdx.x;
    const int lane   = tid & 31;
    const int w      = tid >> 5;        // wave id 0..7 (wave32)
    const int lane16 = lane & 15;
    const int lanehi = lane >> 4;       // 0: lanes 0-15, 1: lanes 16-31
    const int m      = blockIdx.x;      // 0..2047
    const int b      = m >> 9;          // m / N
    const int n      = m & (ZN - 1);

    float* sX    = smem + OFF_X;
    float* sQ    = smem + OFF_Q;
    float* sK    = smem + OFF_K;
    float* sV    = smem + OFF_V;
    float* sY    = smem + OFF_Y;
    float* sW    = smem + OFF_W;        // [3][64*ST]
    float* sAttn = smem + OFF_ATTN;
    float* sBias = smem + OFF_CONST;    // [3][64]
    float* sGam  = sBias + 3*ZH;
    float* sBet  = sGam + ZH;

    // ---------------- phase 0: stage inputs into LDS -----------------------
    // traffic/text rows for this (b, n): 128 rows x 64 f32 each = 2048 float4
#pragma unroll
    for (int i = 0; i < 8; ++i) {
        int idx4 = tid + i * 256;
        int row  = idx4 >> 4;           // 16 float4 per row
        int c4   = idx4 & 15;
        int goff = (((b * ZS + row) * ZN + n) << 6) + (c4 << 2);
        *(float4*)(sX + row * ST + (c4 << 2)) = *(const float4*)(traffic + goff);
        *(float4*)(sY + row * ST + (c4 << 2)) = *(const float4*)(text + goff);
    }
    // weights: 3 matrices x 1024 float4
#pragma unroll
    for (int i = 0; i < 12; ++i) {
        int idx4 = tid + i * 256;       // 0..3071
        int p    = i >> 2;              // uniform per unrolled i (256 f4 steps, 1024 per matrix)
        int r    = (idx4 >> 4) & 63;
        int c4   = idx4 & 15;
        const float* wp = (p == 0) ? Wq : (p == 1) ? Wk : Wv;
        *(float4*)(sW + p * ZH * ST + r * ST + (c4 << 2)) =
            *(const float4*)(wp + (r << 6) + (c4 << 2));
    }
    if (tid < ZH) {
        sBias[tid]          = bq[tid];
        sBias[ZH + tid]     = bk[tid];
        sBias[2 * ZH + tid] = bv[tid];
        sGam[tid]           = gamma[tid];
        sBet[tid]           = beta[tid];
    }
    __syncthreads();

    // ---------------- phase 1: Q/K/V projections (WMMA f32 16x16x4) --------
    // wave w computes output rows [16w, 16w+16); y = x @ W^T + b, so
    // B-matrix element B[k=h][n=o] = W[o][h] (a row of W read along h).
    {
        const int arow = 16 * w + lane16;
#pragma unroll
        for (int p = 0; p < 3; ++p) {
            const float* sA = (p == 0) ? sX : sY;
            const float* sB = sW + p * ZH * ST;
            float* sD       = (p == 0) ? sQ : (p == 1) ? sK : sV;
            const float* bias = sBias + p * ZH;
#pragma unroll
            for (int ct = 0; ct < 4; ++ct) {
                v8f acc = {0.f, 0.f, 0.f, 0.f, 0.f, 0.f, 0.f, 0.f};
#pragma unroll
                for (int kk = 0; kk < 16; ++kk) {
                    int k0 = kk * 4 + 2 * lanehi;
                    v2f a  = *(const v2f*)(sA + arow * ST + k0);
                    v2f bb = *(const v2f*)(sB + (ct * 16 + lane16) * ST + k0);
                    acc = __builtin_amdgcn_wmma_f32_16x16x4_f32(
                        false, a, false, bb, (short)0, acc, false, false);
                }
                float bval = bias[ct * 16 + lane16];
#pragma unroll
                for (int r = 0; r < 8; ++r) {
                    int row = 16 * w + r + 8 * lanehi;
                    sD[row * ST + ct * 16 + lane16] = acc[r] + bval;
                }
            }
        }
    }
    __syncthreads();

    // ---------------- phase 2: scores = q @ k^T, rows [16w,16w+16) ---------
    v8f sc[8];
    {
        const int arow = 16 * w + lane16;
#pragma unroll
        for (int ct = 0; ct < 8; ++ct) {
            v8f acc = {0.f, 0.f, 0.f, 0.f, 0.f, 0.f, 0.f, 0.f};
#pragma unroll
            for (int kk = 0; kk < 16; ++kk) {
                int k0 = kk * 4 + 2 * lanehi;
                v2f a  = *(const v2f*)(sQ + arow * ST + k0);
                v2f bb = *(const v2f*)(sK + (ct * 16 + lane16) * ST + k0);  // B[h][t] = k[t][h]
                acc = __builtin_amdgcn_wmma_f32_16x16x4_f32(
                    false, a, false, bb, (short)0, acc, false, false);
            }
            sc[ct] = acc;
        }
    }
    __syncthreads();  // all waves done reading sY/sW before sAttn overwrites region B

    // ---------------- phase 3: per-row top-16 + softmax --------------------
    // Register layout: sc[ct][r] = scores[16w + r + 8*lanehi][16*ct + lane16].
    // Each 16-lane half-wave cooperatively processes one row (2 rows at a time).
#pragma unroll 1
    for (int r = 0; r < 8; ++r) {
        float orig[8], vals[8];
#pragma unroll
        for (int ct = 0; ct < 8; ++ct) {
            orig[ct] = sc[ct][r] * 0.125f;   // 1/sqrt(H)
            vals[ct] = orig[ct];
        }
        float rowmax = 0.f, sum = 0.f;
        unsigned selmask = 0u;
#pragma unroll 1
        for (int it = 0; it < TOPK; ++it) {
            float bvv = -INFINITY;
            int   bc  = 0x7fffffff;
#pragma unroll
            for (int ct = 0; ct < 8; ++ct) {
                float vv = vals[ct];
                int   cc = ct * 16 + lane16;
                if (vv > bvv || (vv == bvv && cc < bc)) { bvv = vv; bc = cc; }
            }
#pragma unroll
            for (int off = 1; off < 16; off <<= 1) {   // stays within 16-lane group
                float ov = __shfl_xor(bvv, off, 32);
                int   oc = __shfl_xor(bc, off, 32);
                if (ov > bvv || (ov == bvv && oc < bc)) { bvv = ov; bc = oc; }
            }
            if (it == 0) rowmax = bvv;
            sum += __expf(bvv - rowmax);
            int bct = bc >> 4;
            if ((bc & 15) == lane16) {                 // remove selected element
                vals[bct] = -INFINITY;
                selmask |= (1u << bct);
            }
        }
        float inv = 1.0f / sum;
        int row = 16 * w + r + 8 * lanehi;
#pragma unroll
        for (int ct = 0; ct < 8; ++ct) {
            int col = ct * 16 + lane16;
            float p = (selmask >> ct & 1u) ? __expf(orig[ct] - rowmax) * inv : 0.f;
            sAttn[row * STA + col] = p;
            attn_out[(size_t)m * (ZS * ZS) + row * ZS + col] = p;
        }
    }

    // ---------------- phase 4: aligned = attn @ v --------------------------
    // A rows are this wave's own sAttn rows (same-wave LDS ordering suffices);
    // sV was written in phase 1 (covered by the phase-1 barrier).
    v8f al[4];
    {
        const int arow = 16 * w + lane16;
#pragma unroll
        for (int ct = 0; ct < 4; ++ct) {
            v8f acc = {0.f, 0.f, 0.f, 0.f, 0.f, 0.f, 0.f, 0.f};
#pragma unroll
            for (int kk = 0; kk < 32; ++kk) {
                int k0 = kk * 4 + 2 * lanehi;
                v2f a = *(const v2f*)(sAttn + arow * STA + k0);
                v2f bb;
                bb.x = sV[k0 * ST + ct * 16 + lane16];          // B[t][h] = v[t][h]
                bb.y = sV[(k0 + 1) * ST + ct * 16 + lane16];
                acc = __builtin_amdgcn_wmma_f32_16x16x4_f32(
                    false, a, false, bb, (short)0, acc, false, false);
            }
            al[ct] = acc;
        }
    }

    // ---------------- phase 5: residual + LayerNorm + store ----------------
#pragma unroll 1
    for (int r = 0; r < 8; ++r) {
        int row = 16 * w + r + 8 * lanehi;
        float x[4];
        float s1 = 0.f, s2 = 0.f;
#pragma unroll
        for (int ct = 0; ct < 4; ++ct) {
            float t = sX[row * ST + ct * 16 + lane16] + al[ct][r];
            x[ct] = t;
            s1 += t;
            s2 += t * t;
        }
#pragma unroll
        for (int off = 1; off < 16; off <<= 1) {
            s1 += __shfl_xor(s1, off, 32);
            s2 += __shfl_xor(s2, off, 32);
        }
        float mu   = s1 * (1.0f / ZH);
        float var  = s2 * (1.0f / ZH) - mu * mu;
        float rstd = rsqrtf(var + 1e-5f);
#pragma unroll
        for (int ct = 0; ct < 4; ++ct) {
            int col = ct * 16 + lane16;
            float y = (x[ct] - mu) * rstd * sGam[col] + sBet[col];
            out[(size_t)(((b * ZS + row) * ZN + n) << 6) + col] = y;
        }
    }
}

extern "C" void kernel_launch(void* const* d_in, const int* in_sizes, int n_in,
                              void* d_out, int out_size, void* d_ws, size_t ws_size,
                              hipStream_t stream) {
    (void)in_sizes; (void)n_in; (void)out_size; (void)d_ws; (void)ws_size;
    const float* traffic = (const float*)d_in[0];
    const float* text    = (const float*)d_in[1];
    const float* Wq      = (const float*)d_in[2];
    const float* bq      = (const float*)d_in[3];
    const float* Wk      = (const float*)d_in[4];
    const float* bk      = (const float*)d_in[5];
    const float* Wv      = (const float*)d_in[6];
    const float* bv      = (const float*)d_in[7];
    const float* gamma   = (const float*)d_in[8];
    const float* beta    = (const float*)d_in[9];
    // d_in[10] = top_k (fixed 16, baked into the kernel)

    float* out      = (float*)d_out;
    float* attn_out = out + (size_t)ZB * ZS * ZN * ZH;   // outputs concatenated flat

    const size_t shmem = (size_t)LDS_FLOATS * sizeof(float);   // ~222 KB (< 320 KB/WGP)
    sparse_xmodal_fused_kernel<<<ZB * ZN, 256, shmem, stream>>>(
        traffic, text, Wq, bq, Wk, bk, Wv, bv, gamma, beta, out, attn_out);
}